// MatrixNet_32564442038740
// MI455X (gfx1250) — compile-verified
//
#include <hip/hip_runtime.h>

// ---------------------------------------------------------------------------
// MatrixNet on MI455X (gfx1250): braid-matrix network
//   g = x @ (W2*W1)^T  ->  expm(8x8)  ->  chained product over L=64  ->  MLP
// All small-matrix math runs on V_WMMA_F32_16X16X4_F32 with 8x8 channel
// matrices packed block-diagonally (2 channels per 16x16 tile).
// v3: x tile staged with GLOBAL_LOAD_ASYNC_TO_LDS_B32 (ASYNCcnt path) +
//     s_wait_asynccnt fence; rest identical to v2 (transposed coalesced
//     weights, bank-conflict-free LDS padding, prefetch).
// ---------------------------------------------------------------------------

#define BB   2048
#define LL   64
#define INN  14
#define TT   256
#define HCC  256
#define OUTN 9

#define XSTR  17    // x-tile LDS row stride (odd -> conflict-free)
#define SSTR  17    // expm staging row stride
#define SWSZ  (16 * SSTR)  // per-wave staging floats (272)
#define MSTR  257   // MLP LDS row stride

typedef float v2f __attribute__((ext_vector_type(2)));
typedef float v8f __attribute__((ext_vector_type(8)));

__device__ __forceinline__ v8f wmma_f32(v2f a, v2f b, v8f c) {
  // D = A(16x4) * B(4x16) + C(16x16), fp32, wave32
  return __builtin_amdgcn_wmma_f32_16x16x4_f32(
      false, a, false, b, (short)0, c, false, false);
}

__device__ __forceinline__ v8f zero8() {
  v8f z = {0.f, 0.f, 0.f, 0.f, 0.f, 0.f, 0.f, 0.f};
  return z;
}

// ---------------------------------------------------------------------------
// Prep 1: WeffT[i][t] = sum_h W2[t][h] * W1[h][i], stored K-major (16 x 256),
// rows 14..15 zero. Coalesced writes (thread t -> column t).
// ---------------------------------------------------------------------------
__global__ void weff_kernel(const float* __restrict__ W1,
                            const float* __restrict__ W2,
                            float* __restrict__ weffT) {
  const int t = blockIdx.x * blockDim.x + threadIdx.x;
  if (t >= TT) return;
  for (int i = 0; i < 16; ++i) {
    float s = 0.f;
    if (i < INN) {
      for (int h = 0; h < 64; ++h) s += W2[t * 64 + h] * W1[h * INN + i];
    }
    weffT[i * TT + t] = s;
  }
}

// ---------------------------------------------------------------------------
// Prep 2: generic [N][K] -> [K][Npad] transpose with zero padding.
// Makes B-fragment weight loads lane-coalesced (adjacent lanes, adjacent n).
// ---------------------------------------------------------------------------
__global__ void transpose_kernel(const float* __restrict__ src,
                                 float* __restrict__ dst,
                                 int N, int K, int Npad) {
  const int idx = blockIdx.x * blockDim.x + threadIdx.x;
  if (idx >= K * Npad) return;
  const int k = idx / Npad;
  const int n = idx % Npad;
  dst[idx] = (n < N) ? src[n * K + k] : 0.f;
}

// ---------------------------------------------------------------------------
// Kernel B: one workgroup (256 threads = 8 waves) per batch element.
//   Phase 0: async-DMA x[b] (64x14) into padded LDS tile   (ASYNCcnt)
//   Phase 1: g[64][256] = xs @ WeffT                       (WMMA, into LDS)
//   Phase 2: expm of 256 8x8 mats, 2-at-a-time block-diag  (WMMA)
//   Phase 3: 6-level tree product over L=64                (WMMA)
//   Phase 4: write flat[b][256]
// ---------------------------------------------------------------------------
__global__ void braid_kernel(const float* __restrict__ x,
                             const float* __restrict__ weffT,
                             float* __restrict__ flat) {
  extern __shared__ float smem[];
  float* mats  = smem;              // 64*256 floats (g, then expm results)
  float* stage = smem + LL * TT;    // 8 waves * SWSZ floats (also x tile)

  const int b    = blockIdx.x;
  const int tid  = threadIdx.x;
  const int wv   = tid >> 5;
  const int lane = tid & 31;
  const int half = lane >> 4;
  const int l16  = lane & 15;
  float* st = stage + wv * SWSZ;

  // ---- Phase 0: stage x tile into LDS ----
  float* xs = stage;                // 64*XSTR floats, reused later per-wave
  // zero the two pad columns (k = 14, 15)
  for (int idx = tid; idx < LL * 2; idx += 256) {
    const int row = idx >> 1;
    xs[row * XSTR + INN + (idx & 1)] = 0.f;
  }
  // async global -> LDS copy of the 64x14 valid region (no VGPR data path;
  // low 32 bits of the flat LDS-aperture address are the LDS byte offset)
  {
    const float* xb = x + (size_t)b * LL * INN;
    for (int idx = tid; idx < LL * INN; idx += 256) {
      const int row = idx / INN;
      const int k   = idx - row * INN;
      const unsigned ldsOff = (unsigned)(size_t)(&xs[row * XSTR + k]);
      const float* gp = xb + idx;
      asm volatile("global_load_async_to_lds_b32 %0, %1, off"
                   :: "v"(ldsOff), "v"(gp)
                   : "memory");
    }
    asm volatile("s_wait_asynccnt 0x0" ::: "memory");
  }
  __syncthreads();

  // ---- Phase 1: g GEMM (M=64 rows l, N=256 cols t, K=16) ----
  for (int tile = wv; tile < 64; tile += 8) {
    const int mt = tile >> 4;   // M-tile (over l)
    const int nt = tile & 15;   // N-tile (over t)
    v8f acc = zero8();
#pragma unroll
    for (int q = 0; q < 4; ++q) {
      v2f a, bf;
#pragma unroll
      for (int j = 0; j < 2; ++j) {
        const int k = q * 4 + j + 2 * half;
        a[j]  = xs[(mt * 16 + l16) * XSTR + k];
        bf[j] = weffT[k * TT + nt * 16 + l16];   // lane-coalesced
      }
      acc = wmma_f32(a, bf, acc);
    }
#pragma unroll
    for (int r = 0; r < 8; ++r)
      mats[(mt * 16 + r + 8 * half) * TT + nt * 16 + l16] = acc[r];
  }
  __syncthreads();

  // ---- Phase 2: expm via scaling(1/32) + order-6 Taylor + 5 squarings ----
  const float inv32 = 1.0f / 32.0f;
  for (int item = wv; item < 128; item += 8) {
    const int l  = item >> 1;
    const int cp = item & 1;               // channel pair (0,1) or (2,3)
    const float* G = mats + l * TT + cp * 128;

    // X fragments (block-diag of two 8x8 channels), both A and B layout
    v2f Xa[4], Xb[4];
#pragma unroll
    for (int q = 0; q < 4; ++q) {
#pragma unroll
      for (int j = 0; j < 2; ++j) {
        const int k = q * 4 + j + 2 * half;
        const bool dA = ((l16 < 8) == (k < 8));
        Xa[q][j] = dA ? G[((l16 >> 3) << 6) + ((l16 & 7) << 3) + (k & 7)] * inv32 : 0.f;
        Xb[q][j] = dA ? G[((k >> 3) << 6) + ((k & 7) << 3) + (l16 & 7)] * inv32 : 0.f;
      }
    }

    // Horner: S = I; for k=6..1: S = I + (X*S)/k
    v2f Sb[4];
#pragma unroll
    for (int q = 0; q < 4; ++q)
#pragma unroll
      for (int j = 0; j < 2; ++j)
        Sb[q][j] = ((q * 4 + j + 2 * half) == l16) ? 1.f : 0.f;

    for (int kk = 6; kk >= 1; --kk) {
      v8f t8 = zero8();
#pragma unroll
      for (int q = 0; q < 4; ++q) t8 = wmma_f32(Xa[q], Sb[q], t8);
      const float rk = 1.0f / (float)kk;
      __builtin_amdgcn_wave_barrier();
#pragma unroll
      for (int r = 0; r < 8; ++r) {
        const int Mr = r + 8 * half;
        st[Mr * SSTR + l16] = t8[r] * rk + ((Mr == l16) ? 1.f : 0.f);
      }
      __builtin_amdgcn_wave_barrier();
#pragma unroll
      for (int q = 0; q < 4; ++q)
#pragma unroll
        for (int j = 0; j < 2; ++j)
          Sb[q][j] = st[(q * 4 + j + 2 * half) * SSTR + l16];
    }

    // 5 squarings: R = R*R (R staged in LDS tile)
    for (int sq = 0; sq < 5; ++sq) {
      v2f Ra[4], Rb[4];
#pragma unroll
      for (int q = 0; q < 4; ++q)
#pragma unroll
        for (int j = 0; j < 2; ++j) {
          const int k = q * 4 + j + 2 * half;
          Ra[q][j] = st[l16 * SSTR + k];
          Rb[q][j] = st[k * SSTR + l16];
        }
      v8f t8 = zero8();
#pragma unroll
      for (int q = 0; q < 4; ++q) t8 = wmma_f32(Ra[q], Rb[q], t8);
      __builtin_amdgcn_wave_barrier();
#pragma unroll
      for (int r = 0; r < 8; ++r) st[(r + 8 * half) * SSTR + l16] = t8[r];
      __builtin_amdgcn_wave_barrier();
    }

    // copy the two diagonal 8x8 blocks back over g (in place for this item)
    float* Gw = mats + l * TT + cp * 128;
    for (int idx = lane; idx < 128; idx += 32) {
      const int blk = idx >> 6;
      const int rr  = (idx >> 3) & 7;
      const int cc  = idx & 7;
      Gw[idx] = st[(blk * 8 + rr) * SSTR + (blk * 8 + cc)];
    }
    __builtin_amdgcn_wave_barrier();
  }
  __syncthreads();

  // ---- Phase 3: tree-reduce chained product over L ----
  for (int level = 0; level < 6; ++level) {
    const int stride = 1 << level;
    const int nItems = (LL >> (level + 1)) * 2;   // pairs * channel-pairs
    for (int item = wv; item < nItems; item += 8) {
      const int p  = item >> 1;
      const int cp = item & 1;
      const int lA = p * 2 * stride;
      const float* GA = mats + lA * TT + cp * 128;
      const float* GB = mats + (lA + stride) * TT + cp * 128;
      v2f Af[4], Bf[4];
#pragma unroll
      for (int q = 0; q < 4; ++q)
#pragma unroll
        for (int j = 0; j < 2; ++j) {
          const int k = q * 4 + j + 2 * half;
          const bool d = ((l16 < 8) == (k < 8));
          Af[q][j] = d ? GA[((l16 >> 3) << 6) + ((l16 & 7) << 3) + (k & 7)] : 0.f;
          Bf[q][j] = d ? GB[((k >> 3) << 6) + ((k & 7) << 3) + (l16 & 7)] : 0.f;
        }
      v8f t8 = zero8();
#pragma unroll
      for (int q = 0; q < 4; ++q) t8 = wmma_f32(Af[q], Bf[q], t8);
      float* GW = mats + lA * TT + cp * 128;
#pragma unroll
      for (int r = 0; r < 8; ++r) {
        const int Mr = r + 8 * half;
        if ((Mr < 8) == (l16 < 8))
          GW[((Mr >> 3) << 6) + ((Mr & 7) << 3) + (l16 & 7)] = t8[r];
      }
    }
    __syncthreads();
  }

  // ---- Phase 4: flat[b][t] ----
  flat[(size_t)b * TT + tid] = mats[tid];
}

// ---------------------------------------------------------------------------
// Kernel C: fused MLP head. One block (128 threads = 4 waves) per 16 rows.
// Weights pre-transposed to [K][N] so B fragments are lane-coalesced.
// ---------------------------------------------------------------------------
__global__ void mlp_kernel(const float* __restrict__ flat,
                           const float* __restrict__ l1wT, const float* __restrict__ l1b,
                           const float* __restrict__ l2wT, const float* __restrict__ l2b,
                           const float* __restrict__ owT,  const float* __restrict__ ob,
                           float* __restrict__ out) {
  __shared__ float inA[16 * MSTR];
  __shared__ float tmp[16 * MSTR];
  const int m0   = blockIdx.x * 16;
  const int tid  = threadIdx.x;
  const int wv   = tid >> 5;
  const int lane = tid & 31;
  const int half = lane >> 4;
  const int l16  = lane & 15;

  for (int idx = tid; idx < 16 * HCC; idx += 128) {
    const int row = idx >> 8;
    const int col = idx & 255;
    inA[row * MSTR + col] = flat[(size_t)m0 * HCC + idx];
  }
  __syncthreads();

  // layer 1: inA -> tmp
#pragma unroll
  for (int nti = 0; nti < 4; ++nti) {
    const int nt = wv * 4 + nti;
    const int n  = nt * 16 + l16;
    v8f acc = zero8();
    for (int q = 0; q < 64; ++q) {
      if (q + 8 < 64)
        __builtin_prefetch(&l1wT[(size_t)(q * 4 + 32) * HCC + n], 0, 1);
      v2f a, bf;
#pragma unroll
      for (int j = 0; j < 2; ++j) {
        const int k = q * 4 + j + 2 * half;
        a[j]  = inA[l16 * MSTR + k];
        bf[j] = l1wT[(size_t)k * HCC + n];     // lane-coalesced
      }
      acc = wmma_f32(a, bf, acc);
    }
    const float bias = l1b[n];
#pragma unroll
    for (int r = 0; r < 8; ++r) {
      const float v = acc[r] + bias;
      tmp[(r + 8 * half) * MSTR + n] = v > 0.f ? v : 0.f;
    }
  }
  __syncthreads();

  // layer 2: tmp -> inA
#pragma unroll
  for (int nti = 0; nti < 4; ++nti) {
    const int nt = wv * 4 + nti;
    const int n  = nt * 16 + l16;
    v8f acc = zero8();
    for (int q = 0; q < 64; ++q) {
      if (q + 8 < 64)
        __builtin_prefetch(&l2wT[(size_t)(q * 4 + 32) * HCC + n], 0, 1);
      v2f a, bf;
#pragma unroll
      for (int j = 0; j < 2; ++j) {
        const int k = q * 4 + j + 2 * half;
        a[j]  = tmp[l16 * MSTR + k];
        bf[j] = l2wT[(size_t)k * HCC + n];     // lane-coalesced
      }
      acc = wmma_f32(a, bf, acc);
    }
    const float bias = l2b[n];
#pragma unroll
    for (int r = 0; r < 8; ++r) {
      const float v = acc[r] + bias;
      inA[(r + 8 * half) * MSTR + n] = v > 0.f ? v : 0.f;
    }
  }
  __syncthreads();

  // output layer (N=9 zero-padded to 16 in owT), wave 0 only (full wave EXEC)
  if (wv == 0) {
    v8f acc = zero8();
    for (int q = 0; q < 64; ++q) {
      v2f a, bf;
#pragma unroll
      for (int j = 0; j < 2; ++j) {
        const int k = q * 4 + j + 2 * half;
        a[j]  = inA[l16 * MSTR + k];
        bf[j] = owT[k * 16 + l16];             // zero-padded, no divergence
      }
      acc = wmma_f32(a, bf, acc);
    }
    if (l16 < OUTN) {
      const float bias = ob[l16];
#pragma unroll
      for (int r = 0; r < 8; ++r)
        out[(size_t)(m0 + r + 8 * half) * OUTN + l16] = acc[r] + bias;
    }
  }
}

// ---------------------------------------------------------------------------
extern "C" void kernel_launch(void* const* d_in, const int* in_sizes, int n_in,
                              void* d_out, int out_size, void* d_ws, size_t ws_size,
                              hipStream_t stream) {
  (void)in_sizes; (void)n_in; (void)out_size; (void)ws_size;
  const float* x   = (const float*)d_in[0];
  const float* W1  = (const float*)d_in[1];
  const float* W2  = (const float*)d_in[2];
  const float* l1w = (const float*)d_in[3];
  const float* l1b = (const float*)d_in[4];
  const float* l2w = (const float*)d_in[5];
  const float* l2b = (const float*)d_in[6];
  const float* ow  = (const float*)d_in[7];
  const float* ob  = (const float*)d_in[8];
  float* out = (float*)d_out;

  // workspace layout (floats)
  float* weffT = (float*)d_ws;            // 16*256
  float* l1wT  = weffT + 16 * TT;         // 256*256
  float* l2wT  = l1wT + HCC * HCC;        // 256*256
  float* owT   = l2wT + HCC * HCC;        // 256*16
  float* flat  = owT + HCC * 16;          // 2048*256

  weff_kernel<<<1, 256, 0, stream>>>(W1, W2, weffT);
  transpose_kernel<<<(HCC * HCC) / 256, 256, 0, stream>>>(l1w, l1wT, HCC, HCC, HCC);
  transpose_kernel<<<(HCC * HCC) / 256, 256, 0, stream>>>(l2w, l2wT, HCC, HCC, HCC);
  transpose_kernel<<<(HCC * 16) / 256, 256, 0, stream>>>(ow, owT, OUTN, HCC, 16);

  const size_t smem = (size_t)(LL * TT + 8 * SWSZ) * sizeof(float);
  braid_kernel<<<BB, 256, smem, stream>>>(x, weffT, flat);

  mlp_kernel<<<BB / 16, 128, 0, stream>>>(flat, l1wT, l1b, l2wT, l2b, owT, ob, out);
}